// SortPoolClassifier_77704548319508
// MI455X (gfx1250) — compile-verified
//
#include <hip/hip_runtime.h>
#include <hip/hip_bf16.h>

typedef __attribute__((ext_vector_type(2))) float v2f;
typedef __attribute__((ext_vector_type(8))) float v8f;

#define NB 128          // batch count B
#define HF 128          // hidden width H
#define KTOP 30
#define C_OUT 32
#define KER 5
#define L_OUT 26        // K - KER + 1
#define OUT_DIM 10

__device__ __forceinline__ void atomAddF(float* p, float v) {
    unsafeAtomicAdd(p, v);   // hardware global_atomic_add_f32
}

// ---------------- degree / dinv ----------------
__global__ void deg_init_k(float* deg, int n) {
    int i = blockIdx.x * 256 + threadIdx.x;
    if (i < n) deg[i] = 1.0f;                 // self loop
}
__global__ void deg_scatter_k(const int* __restrict__ col, float* deg, int nE) {
    int e = blockIdx.x * 256 + threadIdx.x;
    if (e < nE) atomAddF(&deg[col[e]], 1.0f);
}
__global__ void dinv_k(float* deg, int n) {
    int i = blockIdx.x * 256 + threadIdx.x;
    if (i < n) deg[i] = rsqrtf(fmaxf(deg[i], 1.0f));
}

// ---------------- fp32 WMMA GEMM: out[N x 128] = A[N x 128] @ W[128 x 128] ----
// block = 256 threads (8 waves); each wave owns a 16-row stripe, full 128 cols.
// W staged in LDS k-pair interleaved: wlds[(k>>1)*256 + n*2 + (k&1)], so each
// B fragment is a single aligned ds_load_b64 into an adjacent VGPR pair.
__global__ void gemm_k(const float* __restrict__ A, const float* __restrict__ W,
                       float* __restrict__ out, int nrows) {
    __shared__ float wlds[HF * HF];           // 64 KB
    int tid = threadIdx.x;
    for (int i = tid; i < (HF / 2) * HF; i += 256) {
        int kp = i >> 7;                      // k-pair 0..63
        int n  = i & 127;
        float2 w2;
        w2.x = W[(2 * kp) * HF + n];
        w2.y = W[(2 * kp + 1) * HF + n];
        ((float2*)wlds)[kp * HF + n] = w2;
    }
    __syncthreads();

    int wave = tid >> 5;
    int lane = tid & 31;
    int half = lane >> 4;                     // K sub-pair selector
    int lm   = lane & 15;
    int rowBase = blockIdx.x * 128 + wave * 16;
    int row  = rowBase + lm;
    int srow = row < nrows ? row : (nrows - 1);   // clamp: dead rows never stored
    const float* arow = A + (long long)srow * HF + 2 * half;

    v8f acc[8] = {};                          // 16x128 f32 accumulator stripe
    v2f a_cur = *(const v2f*)arow;            // k0 = 0 fragment

    for (int k0 = 0; k0 < HF; k0 += 4) {
        // software-pipelined next A fragment (branchless wrap on last iter)
        v2f a_next = *(const v2f*)(arow + ((k0 + 4) & (HF - 1)));
        int kp = (k0 >> 1) + half;            // k-pair row for this lane half
        const float* bbase = wlds + kp * 256 + lm * 2;
#pragma unroll
        for (int j = 0; j < 8; j++) {
            v2f b = *(const v2f*)(bbase + j * 32);   // one ds_load_b64
            acc[j] = __builtin_amdgcn_wmma_f32_16x16x4_f32(
                false, a_cur, false, b, (short)0, acc[j], false, false);
        }
        a_cur = a_next;
    }
    // C/D layout: VGPR v, lanes 0-15 -> M=v, lanes 16-31 -> M=v+8; N=lane%16
#pragma unroll
    for (int j = 0; j < 8; j++) {
#pragma unroll
        for (int v = 0; v < 8; v++) {
            int r = rowBase + v + 8 * half;
            if (r < nrows) out[(long long)r * HF + j * 16 + lm] = acc[j][v];
        }
    }
}

// ---------------- aggregation: out = b + dinv^2*hw (self loop), then edges ---
__global__ void agg_init_k(const float* __restrict__ hw, const float* __restrict__ dinv,
                           const float* __restrict__ bias, float* __restrict__ out, int n) {
    int gid = blockIdx.x * 256 + threadIdx.x;
    if (gid >= n * HF) return;
    int i = gid >> 7, f = gid & 127;
    float d = dinv[i];
    out[gid] = bias[f] + d * d * hw[gid];
}

// one edge per 32 lanes, float4 per lane (512-B coalesced gather, L2-resident)
__global__ void edge_scatter_k(const int* __restrict__ row, const int* __restrict__ col,
                               const float* __restrict__ dinv, const float* __restrict__ hw,
                               float* out, int nE) {
    int gid = blockIdx.x * 256 + threadIdx.x;
    int e = gid >> 5;
    if (e >= nE) return;
    int lane = gid & 31;
    int r = row[e], c = col[e];
    float nrm = dinv[r] * dinv[c];
    float4 m = ((const float4*)(hw + (long long)r * HF))[lane];
    float* dst = out + (long long)c * HF + lane * 4;
    atomAddF(dst + 0, m.x * nrm);
    atomAddF(dst + 1, m.y * nrm);
    atomAddF(dst + 2, m.z * nrm);
    atomAddF(dst + 3, m.w * nrm);
}

__global__ void relu_k(float* x, int n) {
    int i = blockIdx.x * 256 + threadIdx.x;
    if (i < n) x[i] = fmaxf(x[i], 0.0f);
}

// ---------------- per-graph segment bookkeeping ----------------
__global__ void count_init_k(int* counts) { counts[threadIdx.x] = 0; }
__global__ void count_scatter_k(const int* __restrict__ batch, int* counts, int n) {
    int i = blockIdx.x * 256 + threadIdx.x;
    if (i < n) atomicAdd(&counts[batch[i]], 1);
}
__global__ void starts_k(const int* __restrict__ counts, int* starts) {
    if (threadIdx.x == 0) {
        int s = 0;
        for (int b = 0; b < NB; b++) { starts[b] = s; s += counts[b]; }
    }
}

// ---------------- sort-pool: top-K by last feature, descending, stable ------
__global__ void sortpool_k(const float* __restrict__ h, const int* __restrict__ counts,
                           const int* __restrict__ starts, float* __restrict__ pooled) {
    __shared__ float smax[256];
    __shared__ int   sidx[256];
    __shared__ int   sel[KTOP];
    int b = blockIdx.x, tid = threadIdx.x;
    int start = starts[b], cnt = counts[b];
    int kk = cnt < KTOP ? cnt : KTOP;

    float lastVal = __builtin_inff();
    int   lastIdx = -1;
    for (int k = 0; k < kk; k++) {
        float bv = -__builtin_inff();
        int   bi = 0x7fffffff;
        for (int i = start + tid; i < start + cnt; i += 256) {
            float v = h[(long long)i * HF + (HF - 1)];
            bool elig = (v < lastVal) || (v == lastVal && i > lastIdx);
            if (elig && (v > bv || (v == bv && i < bi))) { bv = v; bi = i; }
        }
        smax[tid] = bv; sidx[tid] = bi;
        __syncthreads();
        for (int sft = 128; sft > 0; sft >>= 1) {
            if (tid < sft) {
                float ov = smax[tid + sft]; int oi = sidx[tid + sft];
                if (ov > smax[tid] || (ov == smax[tid] && oi < sidx[tid])) {
                    smax[tid] = ov; sidx[tid] = oi;
                }
            }
            __syncthreads();
        }
        if (tid == 0) sel[k] = sidx[0];
        lastVal = smax[0]; lastIdx = sidx[0];
        __syncthreads();
    }
    for (int k = 0; k < KTOP; k++) {
        float* dst = pooled + ((long long)b * KTOP + k) * HF;
        if (k < kk) {
            int node = sel[k];
            for (int f = tid; f < HF; f += 256) dst[f] = h[(long long)node * HF + f];
        } else {
            for (int f = tid; f < HF; f += 256) dst[f] = 0.0f;
        }
    }
}

// ---------------- conv1d (NCH, 128->32 ch, k=5) + relu + flatten ------------
__global__ void conv_k(const float* __restrict__ pooled, const float* __restrict__ cw,
                       const float* __restrict__ cb, float* __restrict__ out) {
    __shared__ float p[KTOP * HF];            // 15 KB
    int b = blockIdx.x, tid = threadIdx.x;
    for (int i = tid; i < KTOP * HF; i += 256) p[i] = pooled[(long long)b * KTOP * HF + i];
    __syncthreads();
    for (int o = tid; o < C_OUT * L_OUT; o += 256) {
        int c = o / L_OUT, l = o % L_OUT;
        float s = cb[c];
        const float* w = cw + c * HF * KER;
        for (int ch = 0; ch < HF; ch++) {
            const float* wc = w + ch * KER;
#pragma unroll
            for (int t = 0; t < KER; t++) s += p[(l + t) * HF + ch] * wc[t];
        }
        out[(long long)b * (C_OUT * L_OUT) + o] = fmaxf(s, 0.0f);
    }
}

// ---------------- lin1 (832->128) + relu ----------------
__global__ void lin1_k(const float* __restrict__ xin, const float* __restrict__ w,
                       const float* __restrict__ bias, float* __restrict__ out) {
    __shared__ float xr[C_OUT * L_OUT];
    int b = blockIdx.x, tid = threadIdx.x;   // block = 128
    for (int i = tid; i < C_OUT * L_OUT; i += 128) xr[i] = xin[(long long)b * (C_OUT * L_OUT) + i];
    __syncthreads();
    float s = bias[tid];
    for (int i = 0; i < C_OUT * L_OUT; i++) s += xr[i] * w[i * HF + tid];
    out[b * HF + tid] = fmaxf(s, 0.0f);
}

// ---------------- lin2 (128->10) ----------------
__global__ void lin2_k(const float* __restrict__ xin, const float* __restrict__ w,
                       const float* __restrict__ bias, float* __restrict__ out) {
    __shared__ float xr[HF];
    int b = blockIdx.x, tid = threadIdx.x;   // block = 128
    xr[tid] = xin[b * HF + tid];
    __syncthreads();
    if (tid < OUT_DIM) {
        float s = bias[tid];
        for (int i = 0; i < HF; i++) s += xr[i] * w[i * OUT_DIM + tid];
        out[b * OUT_DIM + tid] = s;
    }
}

extern "C" void kernel_launch(void* const* d_in, const int* in_sizes, int n_in,
                              void* d_out, int out_size, void* d_ws, size_t ws_size,
                              hipStream_t stream) {
    const float* x      = (const float*)d_in[0];
    const int*   ei     = (const int*)d_in[1];
    const int*   batch  = (const int*)d_in[2];
    const float* W0 = (const float*)d_in[3];  const float* b0 = (const float*)d_in[4];
    const float* W1 = (const float*)d_in[5];  const float* b1 = (const float*)d_in[6];
    const float* W2 = (const float*)d_in[7];  const float* b2 = (const float*)d_in[8];
    const float* cw = (const float*)d_in[9];  const float* cb = (const float*)d_in[10];
    const float* l1w = (const float*)d_in[11]; const float* l1b = (const float*)d_in[12];
    const float* l2w = (const float*)d_in[13]; const float* l2b = (const float*)d_in[14];

    const int N = in_sizes[0] / HF;           // 50000
    const int E = in_sizes[1] / 2;            // 600000
    const int* rowp = ei;                     // edge_index[0]
    const int* colp = ei + E;                 // edge_index[1]

    // workspace carve-up (256-B aligned)
    size_t off = 0;
    auto carve = [&](size_t bytes) {
        void* p = (char*)d_ws + off;
        off += (bytes + 255) & ~(size_t)255;
        return p;
    };
    float* dinv   = (float*)carve((size_t)N * 4);
    float* hw     = (float*)carve((size_t)N * HF * 4);
    float* hA     = (float*)carve((size_t)N * HF * 4);
    int*   counts = (int*)carve(NB * 4);
    int*   starts = (int*)carve(NB * 4);
    float* pooled = (float*)carve((size_t)NB * KTOP * HF * 4);
    float* convo  = (float*)carve((size_t)NB * C_OUT * L_OUT * 4);
    float* l1o    = (float*)carve((size_t)NB * HF * 4);
    (void)ws_size;

    const int TB = 256;
    int gN    = (N + TB - 1) / TB;
    int gE    = (E + TB - 1) / TB;
    int gNH   = (N * HF + TB - 1) / TB;
    int gE32  = (E * 32 + TB - 1) / TB;
    int gGemm = (N + 127) / 128;

    // degrees -> dinv
    deg_init_k<<<gN, TB, 0, stream>>>(dinv, N);
    deg_scatter_k<<<gE, TB, 0, stream>>>(colp, dinv, E);
    dinv_k<<<gN, TB, 0, stream>>>(dinv, N);

    // GCN layer 1: x -> hA
    gemm_k<<<gGemm, TB, 0, stream>>>(x, W0, hw, N);
    agg_init_k<<<gNH, TB, 0, stream>>>(hw, dinv, b0, hA, N);
    edge_scatter_k<<<gE32, TB, 0, stream>>>(rowp, colp, dinv, hw, hA, E);
    relu_k<<<gNH, TB, 0, stream>>>(hA, N * HF);

    // GCN layer 2: hA -> hA (hw intermediary)
    gemm_k<<<gGemm, TB, 0, stream>>>(hA, W1, hw, N);
    agg_init_k<<<gNH, TB, 0, stream>>>(hw, dinv, b1, hA, N);
    edge_scatter_k<<<gE32, TB, 0, stream>>>(rowp, colp, dinv, hw, hA, E);
    relu_k<<<gNH, TB, 0, stream>>>(hA, N * HF);

    // GCN layer 3
    gemm_k<<<gGemm, TB, 0, stream>>>(hA, W2, hw, N);
    agg_init_k<<<gNH, TB, 0, stream>>>(hw, dinv, b2, hA, N);
    edge_scatter_k<<<gE32, TB, 0, stream>>>(rowp, colp, dinv, hw, hA, E);
    relu_k<<<gNH, TB, 0, stream>>>(hA, N * HF);

    // per-graph segments (batch is sorted -> contiguous ranges)
    count_init_k<<<1, NB, 0, stream>>>(counts);
    count_scatter_k<<<gN, TB, 0, stream>>>(batch, counts, N);
    starts_k<<<1, 32, 0, stream>>>(counts, starts);

    // sort-pool + head
    sortpool_k<<<NB, TB, 0, stream>>>(hA, counts, starts, pooled);
    conv_k<<<NB, TB, 0, stream>>>(pooled, cw, cb, convo);
    lin1_k<<<NB, HF, 0, stream>>>(convo, l1w, l1b, l1o);
    lin2_k<<<NB, HF, 0, stream>>>(l1o, l2w, l2b, (float*)d_out);
}